// Mamba2_81037442941633
// MI455X (gfx1250) — compile-verified
//
#include <hip/hip_runtime.h>
#include <hip/hip_bf16.h>

// ---------------------------------------------------------------------------
// Mamba-2 forward for MI455X (gfx1250), fp32 end-to-end.
// - All GEMM-shaped einsums run on V_WMMA_F32_16X16X4_F32 (wave32 matrix pipe)
// - Projection GEMMs stage tiles in LDS via GLOBAL_LOAD_ASYNC_TO_LDS_B128
//   (ASYNCcnt path) for block-level reuse: ~6.7 GB -> ~2.8 GB cache traffic.
// Problem is HBM-bound (~70 GFLOP over ~60 MB unique @ 23.3 TB/s), so fp32
// WMMA (reference-exact) is the right precision.
// ---------------------------------------------------------------------------

typedef float v2f __attribute__((ext_vector_type(2)));
typedef float v8f __attribute__((ext_vector_type(8)));

#define B_SZ    2
#define L_SZ    2048
#define DMODEL  1024
#define DINNER  2048
#define NHEADS  32
#define HEADDIM 64
#define DSTATE  128
#define NCHUNK  8
#define CHUNKT  256
#define DPROJ   4384          // 2*DINNER + 2*DSTATE + NHEADS
#define CONVDIM 2304          // DINNER + 2*DSTATE
#define ROWS    (B_SZ * L_SZ) // 4096

__device__ __forceinline__ v8f wmma4(v2f a, v2f b, v8f c) {
  // D = A(16x4 f32) * B(4x16 f32) + C(16x16 f32)
  return __builtin_amdgcn_wmma_f32_16x16x4_f32(
      false, a, false, b, (short)0, c, false, false);
}

// Raw LDS byte offset of a __shared__ object (AS(3) pointer value == offset).
__device__ __forceinline__ unsigned lds_offset(const void* p) {
  return (unsigned)(unsigned long long)(__attribute__((address_space(3))) const void*)p;
}

// Per-lane async copy of 16 bytes global -> LDS (tracked by ASYNCcnt).
__device__ __forceinline__ void async_b128(unsigned ldsOff, const float* g) {
  asm volatile("global_load_async_to_lds_b128 %0, %1, off"
               :: "v"(ldsOff), "v"((unsigned long long)(size_t)g)
               : "memory");
}
__device__ __forceinline__ void wait_async0() {
  asm volatile("s_wait_asynccnt 0" ::: "memory");
}

// ---------------------------------------------------------------------------
// Tiled D[M,N] = A[M,K] @ B[N,K]^T (row-major A,B), fp32 WMMA, LDS-staged.
// Block = 256 threads (8 waves). Block tile 128(M) x 32(N), K sliced by 32.
// Wave w computes rows [128*by + 16w, +16) x all 32 N columns.
// Requires M%128==0, N%32==0, K%32==0 (holds for 4096x4384x1024, 4096x1024x2048).
// ---------------------------------------------------------------------------
#define BM   128
#define BN   32
#define BK   32
#define LDSW 36   // padded LDS row stride (floats): 144B = 16B-aligned rows,
                  // and r*36 mod 64 hits 16 distinct banks for 16 lanes.

__global__ __launch_bounds__(256)
void wmma_gemm_tiled(const float* __restrict__ A, const float* __restrict__ B,
                     float* __restrict__ D, int M, int N, int K) {
  __shared__ float sA[BM * LDSW];
  __shared__ float sB[BN * LDSW];
  const int tid  = threadIdx.x;
  const int lane = tid & 31, wave = tid >> 5;
  const int mBase = blockIdx.y * BM;
  const int nBase = blockIdx.x * BN;
  const int r  = lane & 15;
  const int ks = (lane >> 4) << 1;           // 0 or 2 (K sub-index)
  const int hi = lane >> 4, col = lane & 15;
  const unsigned sAoff = lds_offset(sA);
  const unsigned sBoff = lds_offset(sB);
  const int m0w = wave << 4;                 // wave's m-subtile in block tile

  v8f acc0 = {}, acc1 = {};
  for (int k0 = 0; k0 < K; k0 += BK) {
    __syncthreads();                         // previous stage fully consumed
    // A tile: 128x32 floats = 1024 b128 quads -> 4 per thread
#pragma unroll
    for (int i = 0; i < 4; ++i) {
      const int q   = tid + (i << 8);
      const int row = q >> 3;                // 8 quads per 32-float row
      const int qk  = (q & 7) << 2;
      async_b128(sAoff + (unsigned)(row * LDSW + qk) * 4u,
                 A + (size_t)(mBase + row) * K + k0 + qk);
    }
    // B tile: 32x32 floats = 256 quads -> 1 per thread
    {
      const int row = tid >> 3;
      const int qk  = (tid & 7) << 2;
      async_b128(sBoff + (unsigned)(row * LDSW + qk) * 4u,
                 B + (size_t)(nBase + row) * K + k0 + qk);
    }
    wait_async0();                           // my copies landed in LDS
    __syncthreads();                         // everyone's copies visible
#pragma unroll
    for (int kk = 0; kk < BK; kk += 4) {
      v2f a  = *(const v2f*)&sA[(m0w + r) * LDSW + kk + ks];
      v2f b0 = *(const v2f*)&sB[(r)       * LDSW + kk + ks];
      v2f b1 = *(const v2f*)&sB[(16 + r)  * LDSW + kk + ks];
      acc0 = wmma4(a, b0, acc0);
      acc1 = wmma4(a, b1, acc1);
    }
  }
  float* dp = D + (size_t)(mBase + m0w + 8 * hi) * N + nBase + col;
#pragma unroll
  for (int rr = 0; rr < 8; ++rr) {
    dp[(size_t)rr * N]      = acc0[rr];
    dp[(size_t)rr * N + 16] = acc1[rr];
  }
}

// ---------------------------------------------------------------------------
// Depthwise causal conv (K=4) + SiLU over the xBC slice of zxbcdt.
// ---------------------------------------------------------------------------
__global__ void conv_silu_kernel(const float* __restrict__ zx,
                                 const float* __restrict__ cw,
                                 const float* __restrict__ cb,
                                 float* __restrict__ xbc) {
  int idx = blockIdx.x * blockDim.x + threadIdx.x;
  if (idx >= ROWS * CONVDIM) return;
  const int ch = idx % CONVDIM;
  const int bt = idx / CONVDIM;
  const int t  = bt % L_SZ;
  const int b  = bt / L_SZ;
  float acc = cb[ch];
#pragma unroll
  for (int k = 0; k < 4; ++k) {
    const int tt = t + k - 3;
    if (tt >= 0)
      acc += zx[(size_t)(b * L_SZ + tt) * DPROJ + DINNER + ch] * cw[ch * 4 + k];
  }
  xbc[idx] = acc / (1.0f + expf(-acc));       // SiLU
}

// ---------------------------------------------------------------------------
// dt = softplus(raw + dt_bias); dA = -exp(A_log) * dt
// ---------------------------------------------------------------------------
__global__ void dt_kernel(const float* __restrict__ zx,
                          const float* __restrict__ dt_bias,
                          const float* __restrict__ A_log,
                          float* __restrict__ dtb, float* __restrict__ dAb) {
  int idx = blockIdx.x * blockDim.x + threadIdx.x;
  if (idx >= ROWS * NHEADS) return;
  const int h  = idx & 31;
  const int bt = idx >> 5;
  const float raw = zx[(size_t)bt * DPROJ + (DPROJ - NHEADS) + h] + dt_bias[h];
  const float d   = (raw > 20.0f) ? raw : log1pf(expf(raw));
  dtb[idx] = d;
  dAb[idx] = -expf(A_log[h]) * d;
}

// ---------------------------------------------------------------------------
// Per-(b,h,chunk) inclusive cumsum of dA; also chunk totals.
// Acs layout: [b, h, c, t]; ctot layout: [b, h, c].
// ---------------------------------------------------------------------------
__global__ void cumsum_kernel(const float* __restrict__ dAb,
                              float* __restrict__ Acs, float* __restrict__ ctot) {
  int idx = blockIdx.x * blockDim.x + threadIdx.x;   // (b*32+h)*8 + c
  if (idx >= B_SZ * NHEADS * NCHUNK) return;
  const int c = idx % NCHUNK;
  const int h = (idx / NCHUNK) % NHEADS;
  const int b = idx / (NCHUNK * NHEADS);
  const int t0 = b * L_SZ + c * CHUNKT;
  float s = 0.0f;
  float* out = Acs + (size_t)idx * CHUNKT;
  for (int t = 0; t < CHUNKT; ++t) {
    s += dAb[(size_t)(t0 + t) * NHEADS + h];
    out[t] = s;
  }
  ctot[idx] = s;
}

// ---------------------------------------------------------------------------
// G[bc][256,256] = C_chunk @ B_chunk^T — shared by all 32 heads (NGROUPS=1).
// ---------------------------------------------------------------------------
__global__ void scores_kernel(const float* __restrict__ xbc,
                              float* __restrict__ G) {
  const int lane = threadIdx.x & 31, wave = threadIdx.x >> 5;
  const int bc   = blockIdx.y;                 // b*8+c
  const int tile = blockIdx.x * 8 + wave;      // 0..127
  const int mt = tile >> 3, nt = tile & 7;
  const int m0 = mt << 4, n0 = nt << 5;
  const int r = lane & 15, ks = (lane >> 4) << 1;
  const float* base = xbc + (size_t)bc * CHUNKT * CONVDIM;
  const float* ap  = base + (size_t)(m0 + r) * CONVDIM + (DINNER + DSTATE) + ks; // C
  const float* bp0 = base + (size_t)(n0 + r) * CONVDIM + DINNER + ks;            // B
  const float* bp1 = bp0 + (size_t)16 * CONVDIM;
  v8f acc0 = {}, acc1 = {};
  for (int k0 = 0; k0 < DSTATE; k0 += 4) {
    v2f a  = *(const v2f*)(ap + k0);
    v2f b0 = *(const v2f*)(bp0 + k0);
    v2f b1 = *(const v2f*)(bp1 + k0);
    acc0 = wmma4(a, b0, acc0);
    acc1 = wmma4(a, b1, acc1);
  }
  const int hi = lane >> 4, col = lane & 15;
  float* gp = G + (size_t)bc * CHUNKT * CHUNKT + (size_t)(m0 + 8 * hi) * CHUNKT + n0 + col;
#pragma unroll
  for (int rr = 0; rr < 8; ++rr) {
    gp[rr * CHUNKT]      = acc0[rr];
    gp[rr * CHUNKT + 16] = acc1[rr];
  }
}

// ---------------------------------------------------------------------------
// Per (b, chunk, head):
//   Y_diag[t,p] = sum_{s<=t} G[t,s] * exp(cs[t]-cs[s]) * Xdt[s,p]
//   Sloc[p,n]   = sum_t exp(cs_last - cs[t]) * Xdt[t,p] * B[t,n]
// tiles 0..31 -> Y_diag (16x32), tiles 32..47 -> states (16x32).
// ---------------------------------------------------------------------------
__global__ void ydiag_states_kernel(const float* __restrict__ xbc,
                                    const float* __restrict__ G,
                                    const float* __restrict__ dtb,
                                    const float* __restrict__ Acs,
                                    float* __restrict__ Y,
                                    float* __restrict__ Sloc) {
  __shared__ float cs_s[CHUNKT], dt_s[CHUNKT], dec_s[CHUNKT];
  const int by = blockIdx.y;                  // bc*32 + h
  const int h = by & 31, bc = by >> 5;
  const int tid = threadIdx.x;
  {
    const int aidx = ((bc >> 3) * NHEADS + h) * NCHUNK + (bc & 7);
    cs_s[tid] = Acs[(size_t)aidx * CHUNKT + tid];
    dt_s[tid] = dtb[(size_t)(bc * CHUNKT + tid) * NHEADS + h];
  }
  __syncthreads();
  dec_s[tid] = expf(cs_s[CHUNKT - 1] - cs_s[tid]) * dt_s[tid];
  __syncthreads();

  const int lane = tid & 31, wave = tid >> 5;
  const int tile = blockIdx.x * 8 + wave;     // 0..47
  const int r = lane & 15, ks = (lane >> 4) << 1;
  const int hi = lane >> 4, col = lane & 15;
  const size_t tbase = (size_t)bc * CHUNKT;
  const float* gbase = G + (size_t)bc * CHUNKT * CHUNKT;

  if (tile < 32) {
    const int mt = tile >> 1, pt = tile & 1;
    const int m0 = mt << 4, n0 = pt << 5;
    const int i0 = m0 + r;
    const float csi = cs_s[i0];
    const float* grow = gbase + (size_t)i0 * CHUNKT;
    v8f acc0 = {}, acc1 = {};
    const int kmax = m0 + 16;                 // causal mask: s <= t
    for (int k0 = 0; k0 < kmax; k0 += 4) {
      const int j0 = k0 + ks, j1 = j0 + 1;
      v2f a;
      a.x = (j0 <= i0) ? grow[j0] * expf(csi - cs_s[j0]) : 0.0f;
      a.y = (j1 <= i0) ? grow[j1] * expf(csi - cs_s[j1]) : 0.0f;
      const float* x0 = xbc + (tbase + j0) * CONVDIM + h * HEADDIM;
      const float* x1 = xbc + (tbase + j1) * CONVDIM + h * HEADDIM;
      v2f b0, b1;
      b0.x = x0[n0 + r]      * dt_s[j0];
      b0.y = x1[n0 + r]      * dt_s[j1];
      b1.x = x0[n0 + 16 + r] * dt_s[j0];
      b1.y = x1[n0 + 16 + r] * dt_s[j1];
      acc0 = wmma4(a, b0, acc0);
      acc1 = wmma4(a, b1, acc1);
    }
    float* yp = Y + (tbase + m0 + 8 * hi) * DINNER + h * HEADDIM + n0 + col;
#pragma unroll
    for (int rr = 0; rr < 8; ++rr) {
      yp[(size_t)rr * DINNER]      = acc0[rr];
      yp[(size_t)rr * DINNER + 16] = acc1[rr];
    }
  } else {
    const int st = tile - 32;                 // 0..15
    const int m0 = (st & 3) << 4;             // p tile (HEADDIM)
    const int n0 = (st >> 2) << 5;            // n tile (DSTATE)
    v8f acc0 = {}, acc1 = {};
    for (int k0 = 0; k0 < CHUNKT; k0 += 4) {
      const int j0 = k0 + ks, j1 = j0 + 1;
      const float* x0 = xbc + (tbase + j0) * CONVDIM;
      const float* x1 = xbc + (tbase + j1) * CONVDIM;
      v2f a;                                   // A[p][t] = decay*dt*x[t,p]
      a.x = x0[h * HEADDIM + m0 + r] * dec_s[j0];
      a.y = x1[h * HEADDIM + m0 + r] * dec_s[j1];
      v2f b0, b1;                              // B[t][n]
      b0.x = x0[DINNER + n0 + r];
      b0.y = x1[DINNER + n0 + r];
      b1.x = x0[DINNER + n0 + 16 + r];
      b1.y = x1[DINNER + n0 + 16 + r];
      acc0 = wmma4(a, b0, acc0);
      acc1 = wmma4(a, b1, acc1);
    }
    float* sp = Sloc + (size_t)by * (HEADDIM * DSTATE) +
                (size_t)(m0 + 8 * hi) * DSTATE + n0 + col;
#pragma unroll
    for (int rr = 0; rr < 8; ++rr) {
      sp[rr * DSTATE]      = acc0[rr];
      sp[rr * DSTATE + 16] = acc1[rr];
    }
  }
}

// ---------------------------------------------------------------------------
// Inter-chunk state recurrence: Spre[c] = S; S = exp(tot[c])*S + Sloc[c]
// ---------------------------------------------------------------------------
__global__ void prefix_states_kernel(const float* __restrict__ Sloc,
                                     const float* __restrict__ ctot,
                                     float* __restrict__ Spre) {
  const int bh = blockIdx.x;                  // b*32 + h
  const int b = bh >> 5, h = bh & 31;
  float scale[NCHUNK];
#pragma unroll
  for (int c = 0; c < NCHUNK; ++c) scale[c] = expf(ctot[bh * NCHUNK + c]);
  for (int e = threadIdx.x; e < HEADDIM * DSTATE; e += blockDim.x) {
    float S = 0.0f;
#pragma unroll
    for (int c = 0; c < NCHUNK; ++c) {
      const size_t off =
          ((size_t)((b * NCHUNK + c) * NHEADS + h)) * (HEADDIM * DSTATE) + e;
      Spre[off] = S;
      S = S * scale[c] + Sloc[off];
    }
  }
}

// ---------------------------------------------------------------------------
// Y_off[t,p] = exp(cs[t]) * sum_n C[t,n] * Spre[p,n]; accumulate into Y.
// ---------------------------------------------------------------------------
__global__ void yoff_kernel(const float* __restrict__ xbc,
                            const float* __restrict__ Spre,
                            const float* __restrict__ Acs,
                            float* __restrict__ Y) {
  __shared__ float ecs[CHUNKT];
  const int by = blockIdx.y;                  // bc*32 + h
  const int h = by & 31, bc = by >> 5;
  const int tid = threadIdx.x;
  {
    const int aidx = ((bc >> 3) * NHEADS + h) * NCHUNK + (bc & 7);
    ecs[tid] = expf(Acs[(size_t)aidx * CHUNKT + tid]);
  }
  __syncthreads();
  const int lane = tid & 31, wave = tid >> 5;
  const int tile = blockIdx.x * 8 + wave;     // 0..31
  const int mt = tile >> 1, pt = tile & 1;
  const int m0 = mt << 4, n0 = pt << 5;
  const int r = lane & 15, ks = (lane >> 4) << 1;
  const size_t tbase = (size_t)bc * CHUNKT;
  const float* ap  = xbc + (tbase + m0 + r) * CONVDIM + (DINNER + DSTATE) + ks; // C
  const float* Sp  = Spre + (size_t)by * (HEADDIM * DSTATE);
  const float* bp0 = Sp + (size_t)(n0 + r) * DSTATE + ks;       // B[n][p]=S[p][n]
  const float* bp1 = Sp + (size_t)(n0 + 16 + r) * DSTATE + ks;
  v8f acc0 = {}, acc1 = {};
  for (int k0 = 0; k0 < DSTATE; k0 += 4) {
    v2f a  = *(const v2f*)(ap + k0);
    v2f b0 = *(const v2f*)(bp0 + k0);
    v2f b1 = *(const v2f*)(bp1 + k0);
    acc0 = wmma4(a, b0, acc0);
    acc1 = wmma4(a, b1, acc1);
  }
  const int hi = lane >> 4, col = lane & 15;
  float* yp = Y + (tbase + m0 + 8 * hi) * DINNER + h * HEADDIM + n0 + col;
#pragma unroll
  for (int rr = 0; rr < 8; ++rr) {
    const float e = ecs[m0 + 8 * hi + rr];
    yp[(size_t)rr * DINNER]      += e * acc0[rr];
    yp[(size_t)rr * DINNER + 16] += e * acc1[rr];
  }
}

// ---------------------------------------------------------------------------
// LayerNorm (mean/var over DINNER) * ln_w, gated by SiLU(z). In-place on Y.
// ---------------------------------------------------------------------------
__global__ void ln_gate_kernel(const float* __restrict__ zx,
                               const float* __restrict__ lnw,
                               float* __restrict__ Y) {
  __shared__ float red[256];
  const int row = blockIdx.x;
  const int tid = threadIdx.x;
  float* yr = Y + (size_t)row * DINNER;
  const float* zr = zx + (size_t)row * DPROJ;
  float s = 0.0f, s2 = 0.0f;
  for (int i = tid; i < DINNER; i += 256) {
    const float v = yr[i];
    s += v; s2 += v * v;
  }
  red[tid] = s; __syncthreads();
  for (int o = 128; o > 0; o >>= 1) {
    if (tid < o) red[tid] += red[tid + o];
    __syncthreads();
  }
  const float mu = red[0] * (1.0f / DINNER);
  __syncthreads();
  red[tid] = s2; __syncthreads();
  for (int o = 128; o > 0; o >>= 1) {
    if (tid < o) red[tid] += red[tid + o];
    __syncthreads();
  }
  const float var  = red[0] * (1.0f / DINNER) - mu * mu;
  const float rstd = rsqrtf(var + 1e-5f);
  for (int i = tid; i < DINNER; i += 256) {
    const float v = (yr[i] - mu) * rstd * lnw[i];
    const float z = zr[i];
    yr[i] = v * (z / (1.0f + expf(-z)));
  }
}

// ---------------------------------------------------------------------------
extern "C" void kernel_launch(void* const* d_in, const int* in_sizes, int n_in,
                              void* d_out, int out_size, void* d_ws,
                              size_t ws_size, hipStream_t stream) {
  const float* u          = (const float*)d_in[0];
  const float* in_proj_w  = (const float*)d_in[1];
  const float* conv_w     = (const float*)d_in[2];
  const float* conv_b     = (const float*)d_in[3];
  const float* dt_bias    = (const float*)d_in[4];
  const float* A_log      = (const float*)d_in[5];
  const float* ln_w       = (const float*)d_in[6];
  const float* out_proj_w = (const float*)d_in[7];
  float* out = (float*)d_out;

  // Workspace carve-up (floats). Total ~45.6M floats (~183 MB).
  float* ws      = (float*)d_ws;
  float* zxbcdt  = ws;                                   // ROWS*DPROJ
  float* xbc     = zxbcdt + (size_t)ROWS * DPROJ;        // ROWS*CONVDIM
  float* dtb     = xbc + (size_t)ROWS * CONVDIM;         // ROWS*NHEADS
  float* dAb     = dtb + (size_t)ROWS * NHEADS;          // ROWS*NHEADS
  float* Acs     = dAb + (size_t)ROWS * NHEADS;          // B*H*NC*CHUNK
  float* ctot    = Acs + (size_t)B_SZ * NHEADS * NCHUNK * CHUNKT; // B*H*NC
  float* G       = ctot + (size_t)B_SZ * NHEADS * NCHUNK;         // B*NC*256*256
  float* Sloc    = G + (size_t)B_SZ * NCHUNK * CHUNKT * CHUNKT;   // B*NC*H*64*128
  float* Spre    = Sloc + (size_t)B_SZ * NCHUNK * NHEADS * HEADDIM * DSTATE;
  float* Ybuf    = Spre + (size_t)B_SZ * NCHUNK * NHEADS * HEADDIM * DSTATE;

  // 1) in-proj GEMM: zxbcdt[4096,4384] = u[4096,1024] @ W^T  (LDS async-tiled)
  wmma_gemm_tiled<<<dim3(DPROJ / BN, ROWS / BM), 256, 0, stream>>>(
      u, in_proj_w, zxbcdt, ROWS, DPROJ, DMODEL);
  // 2) conv + SiLU
  conv_silu_kernel<<<(ROWS * CONVDIM) / 256, 256, 0, stream>>>(
      zxbcdt, conv_w, conv_b, xbc);
  // 3) dt / dA
  dt_kernel<<<(ROWS * NHEADS) / 256, 256, 0, stream>>>(
      zxbcdt, dt_bias, A_log, dtb, dAb);
  // 4) chunk cumsums
  cumsum_kernel<<<2, 256, 0, stream>>>(dAb, Acs, ctot);
  // 5) head-shared scores G = C @ B^T per (b,chunk)
  scores_kernel<<<dim3(16, B_SZ * NCHUNK), 256, 0, stream>>>(xbc, G);
  // 6) intra-chunk Y_diag + local states
  ydiag_states_kernel<<<dim3(6, B_SZ * NCHUNK * NHEADS), 256, 0, stream>>>(
      xbc, G, dtb, Acs, Ybuf, Sloc);
  // 7) inter-chunk state recurrence
  prefix_states_kernel<<<B_SZ * NHEADS, 256, 0, stream>>>(Sloc, ctot, Spre);
  // 8) Y_off accumulation
  yoff_kernel<<<dim3(4, B_SZ * NCHUNK * NHEADS), 256, 0, stream>>>(
      xbc, Spre, Acs, Ybuf);
  // 9) LayerNorm + SiLU(z) gate (in-place on Ybuf)
  ln_gate_kernel<<<ROWS, 256, 0, stream>>>(zxbcdt, ln_w, Ybuf);
  // 10) out-proj GEMM: out[4096,1024] = Y[4096,2048] @ Wout^T
  wmma_gemm_tiled<<<dim3(DMODEL / BN, ROWS / BM), 256, 0, stream>>>(
      Ybuf, out_proj_w, out, ROWS, DMODEL, DINNER);
}